// Group_73289321939487
// MI455X (gfx1250) — compile-verified
//
#include <hip/hip_runtime.h>
#include <hip/hip_bf16.h>

typedef __attribute__((ext_vector_type(2))) float v2f;
typedef __attribute__((ext_vector_type(8))) float v8f;
typedef __attribute__((ext_vector_type(4))) unsigned int u32x4;
typedef __attribute__((ext_vector_type(4))) int i32x4;
typedef __attribute__((ext_vector_type(8))) int i32x8;

#define B_    8
#define N_    16384
#define G_    512
#define K_    32
#define NCHUNK 8
#define CHUNK  2048      // N_/NCHUNK
#define GT     16        // centers per WMMA tile

// ---------------------------------------------------------------------------
// Kernel 1: xyz = R * pc + t       (pc: [B,N,3], pose: [B,3,4])
// ---------------------------------------------------------------------------
__global__ void se3_kernel(const float* __restrict__ pc,
                           const float* __restrict__ pose,
                           float* __restrict__ xyz) {
  int gid = blockIdx.x * blockDim.x + threadIdx.x;   // b*N + n
  int b = gid >> 14;
  const float* P = pose + b * 12;
  float x = pc[gid * 3 + 0], y = pc[gid * 3 + 1], z = pc[gid * 3 + 2];
  #pragma unroll
  for (int i = 0; i < 3; ++i) {
    xyz[gid * 3 + i] = fmaf(P[i*4+0], x, fmaf(P[i*4+1], y, fmaf(P[i*4+2], z, P[i*4+3])));
  }
}

// ---------------------------------------------------------------------------
// wave32 argmax / argmin reductions (tie-break: lowest index, as in jnp.argmax)
// ---------------------------------------------------------------------------
__device__ __forceinline__ void wave_argmax(float& v, int& i) {
  #pragma unroll
  for (int off = 16; off > 0; off >>= 1) {
    float ov = __shfl_xor(v, off, 32);
    int   oi = __shfl_xor(i, off, 32);
    if (ov > v || (ov == v && oi < i)) { v = ov; i = oi; }
  }
}
__device__ __forceinline__ void wave_argmin2(float& v, int& i, int& o) {
  #pragma unroll
  for (int off = 16; off > 0; off >>= 1) {
    float ov = __shfl_xor(v, off, 32);
    int   oi = __shfl_xor(i, off, 32);
    int   oo = __shfl_xor(o, off, 32);
    if (ov < v || (ov == v && oi < i)) { v = ov; i = oi; o = oo; }
  }
}

// ---------------------------------------------------------------------------
// Kernel 2: farthest point sampling. One workgroup (1024 thr = 32 waves) per
// batch; 16 points + running dists live in registers per thread.
// ---------------------------------------------------------------------------
__global__ void __launch_bounds__(1024, 1)
fps_kernel(const float* __restrict__ xyz, float* __restrict__ centers) {
  const int b = blockIdx.x;
  const int tid = threadIdx.x;
  const int lane = tid & 31, wave = tid >> 5;

  __shared__ float s_red[32];
  __shared__ int   s_redi[32];
  __shared__ float s_c[3];
  __shared__ int   s_cur;

  float px[16], py[16], pz[16], dist[16];
  #pragma unroll
  for (int i = 0; i < 16; ++i) {
    int n = tid + (i << 10);
    int base = ((b << 14) + n) * 3;
    px[i] = xyz[base + 0];
    py[i] = xyz[base + 1];
    pz[i] = xyz[base + 2];
    dist[i] = 1e10f;
  }

  int cur = 0;
  for (int k = 0; k < G_; ++k) {
    if (tid == (cur & 1023)) {
      int sl = cur >> 10;
      s_c[0] = px[sl]; s_c[1] = py[sl]; s_c[2] = pz[sl];
      centers[(b * G_ + k) * 3 + 0] = px[sl];
      centers[(b * G_ + k) * 3 + 1] = py[sl];
      centers[(b * G_ + k) * 3 + 2] = pz[sl];
    }
    __syncthreads();
    float cx = s_c[0], cy = s_c[1], cz = s_c[2];

    float bv = -1.0f; int bi = 0x7fffffff;
    #pragma unroll
    for (int i = 0; i < 16; ++i) {
      float dx = px[i] - cx, dy = py[i] - cy, dz = pz[i] - cz;
      float d = fmaf(dx, dx, fmaf(dy, dy, dz * dz));
      dist[i] = fminf(dist[i], d);
      if (dist[i] > bv) { bv = dist[i]; bi = tid + (i << 10); }
    }
    wave_argmax(bv, bi);
    if (lane == 0) { s_red[wave] = bv; s_redi[wave] = bi; }
    __syncthreads();
    if (wave == 0) {
      float v = s_red[lane]; int idx = s_redi[lane];
      wave_argmax(v, idx);
      if (lane == 0) s_cur = idx;
    }
    __syncthreads();
    cur = s_cur;
  }
}

// ---------------------------------------------------------------------------
// TDM: stage one contiguous run of `elems` f32 from global into LDS offset 0.
// Built per CDNA5 D# spec (CDNA5 ISA): 1-row tile, data_size=4B.
// ---------------------------------------------------------------------------
__device__ __forceinline__ void tdm_load_f32_to_lds0(const float* gptr, int elems) {
  unsigned long long ga = (unsigned long long)(uintptr_t)gptr;
  u32x4 g0;
  g0.x = 1u;                                            // count=1, user D#
  g0.y = 0u;                                            // lds_addr = 0
  g0.z = (unsigned)(ga & 0xffffffffu);                  // global_addr[31:0]
  g0.w = (unsigned)((ga >> 32) & 0x1ffffffu) | 0x80000000u;  // addr[56:32] | type=2

  i32x8 g1;
  g1[0] = 2 << 16;               // workgroup_mask=0, data_size=4B
  g1[1] = (elems & 0xffff) << 16;        // tensor_dim0[15:0]
  g1[2] = ((elems >> 16) & 0xffff) | (1 << 16);  // tensor_dim0[31:16], tensor_dim1=1
  g1[3] = (elems & 0xffff) << 16;        // tile_dim0
  g1[4] = 1;                     // tile_dim1=1, tile_dim2=0
  g1[5] = elems;                 // tensor_dim0_stride[31:0]
  g1[6] = 0;
  g1[7] = 0;

  i32x4 g2; g2[0] = 1; g2[1] = 0; g2[2] = 0; g2[3] = 0;  // tensor_dim2=1
  i32x4 g3; g3[0] = 0; g3[1] = 0; g3[2] = 0; g3[3] = 0;

#if defined(__clang_major__) && __clang_major__ >= 23
  i32x8 g4; g4[0]=0; g4[1]=0; g4[2]=0; g4[3]=0; g4[4]=0; g4[5]=0; g4[6]=0; g4[7]=0;
  __builtin_amdgcn_tensor_load_to_lds(g0, g1, g2, g3, g4, 0);
#else
  __builtin_amdgcn_tensor_load_to_lds(g0, g1, g2, g3, 0);
#endif
}

// ---------------------------------------------------------------------------
// Kernel 3: fused WMMA distance tile + per-row partial top-32.
// Block = 512 thr (16 waves). Each block: 16 centers x 2048-pt chunk.
// Chunk xyz staged global->LDS by the Tensor Data Mover (TENSORcnt).
// d(g,n) = |x_n|^2 - 2 c_g . x_n   via V_WMMA_F32_16X16X4_F32 with
//   A row m = [-2cx, -2cy, -2cz, 1],  B col n = [x, y, z, |x|^2]
// ---------------------------------------------------------------------------
__global__ void __launch_bounds__(512, 1)
dist_topk_kernel(const float* __restrict__ xyz,
                 const float* __restrict__ centers,
                 float* __restrict__ pvals,
                 int* __restrict__ pidx) {
  extern __shared__ float lds[];
  float* sP = lds;                         // [2048][3] staged points (24KB, offset 0)
  float* sA = lds + CHUNK * 3;             // [16][4]
  float* d2 = lds + CHUNK * 3 + GT * 4;    // [16][2048]

  const int bid = blockIdx.x;
  const int b  = bid / (32 * NCHUNK);
  const int rem = bid % (32 * NCHUNK);
  const int gt = rem / NCHUNK;
  const int ch = rem % NCHUNK;
  const int g0 = gt * GT;
  const int n0 = ch * CHUNK;

  const int tid = threadIdx.x;
  const int lane = tid & 31, wave = tid >> 5;

  // Wave 0 kicks the TDM DMA of this block's point chunk into LDS.
  if (wave == 0) {
    tdm_load_f32_to_lds0(xyz + (size_t)(b * N_ + n0) * 3, CHUNK * 3);
  }
  if (tid < GT) {
    const float* c = centers + ((b * G_ + g0 + tid) * 3);
    sA[tid * 4 + 0] = -2.0f * c[0];
    sA[tid * 4 + 1] = -2.0f * c[1];
    sA[tid * 4 + 2] = -2.0f * c[2];
    sA[tid * 4 + 3] = 1.0f;
  }
  if (wave == 0) __builtin_amdgcn_s_wait_tensorcnt(0);
  __syncthreads();

  // A operand: lanes 0-15 hold (K0,K1) of row m=lane; lanes 16-31 hold (K2,K3)
  const int m = lane & 15;
  const int ksel = lane >> 4;
  v2f a;
  a.x = sA[m * 4 + 2 * ksel + 0];
  a.y = sA[m * 4 + 2 * ksel + 1];
  const int mb = ksel * 8;

  // 128 sub-tiles of 16 points; 16 waves x 8 unrolled tiles per wave.
  #pragma unroll
  for (int it = 0; it < 8; ++it) {
    int col = (wave + it * 16) * 16 + (lane & 15);
    float x = sP[col * 3 + 0];
    float y = sP[col * 3 + 1];
    float z = sP[col * 3 + 2];
    float s = fmaf(x, x, fmaf(y, y, z * z));
    v2f bb;
    bb.x = ksel ? z : x;        // branch-free: v_cndmask, no exec toggling
    bb.y = ksel ? s : y;

    v8f acc = {};
    acc = __builtin_amdgcn_wmma_f32_16x16x4_f32(
        false, a, false, bb, (short)0, acc, false, false);

    #pragma unroll
    for (int v = 0; v < 8; ++v) d2[(mb + v) * CHUNK + col] = acc[v];
  }
  __syncthreads();

  // Per-row partial top-32: wave w selects 32 smallest of its 2048 values.
  const float* row = d2 + wave * CHUNK;
  unsigned long long used = 0ull;
  float* ovals = pvals + (((b * G_ + g0 + wave) * NCHUNK + ch) * K_);
  int*   oidx  = pidx  + (((b * G_ + g0 + wave) * NCHUNK + ch) * K_);

  for (int k = 0; k < K_; ++k) {
    float bv = 3.4e38f; int bc = 0x7fffffff;
    for (int j = 0; j < CHUNK / 32; ++j) {
      int c = lane + (j << 5);
      float v = row[c];
      v = ((used >> j) & 1ull) ? 3.4e38f : v;   // select, not branch
      if (v < bv || (v == bv && c < bc)) { bv = v; bc = c; }
    }
    int dummy = 0;
    wave_argmin2(bv, bc, dummy);
    if ((bc & 31) == lane) used |= 1ull << (bc >> 5);
    if (lane == 0) { ovals[k] = bv; oidx[k] = n0 + bc; }
  }
}

// ---------------------------------------------------------------------------
// Kernel 4: merge NCHUNK partial top-32 lists per center (256 candidates),
// gather xyz, recenter, write neighborhood. One wave per center.
// ---------------------------------------------------------------------------
__global__ void __launch_bounds__(256, 2)
merge_kernel(const float* __restrict__ xyz,
             const float* __restrict__ centers,
             const float* __restrict__ pvals,
             const int* __restrict__ pidx,
             float* __restrict__ out_nb) {
  const int lane = threadIdx.x & 31, wave = threadIdx.x >> 5;
  const int g = blockIdx.x * 8 + wave;     // 0 .. B*G-1
  const int b = g / G_, gg = g % G_;

  const int base = (b * G_ + gg) * (NCHUNK * K_);
  float cv[NCHUNK]; int ci[NCHUNK];
  #pragma unroll
  for (int j = 0; j < NCHUNK; ++j) {
    cv[j] = pvals[base + j * K_ + lane];
    ci[j] = pidx [base + j * K_ + lane];
  }
  float cx = centers[(b * G_ + gg) * 3 + 0];
  float cy = centers[(b * G_ + gg) * 3 + 1];
  float cz = centers[(b * G_ + gg) * 3 + 2];

  unsigned used = 0u;
  for (int k = 0; k < K_; ++k) {
    float bv = 3.4e38f; int bidx = 0x7fffffff; int bslot = 0;
    #pragma unroll
    for (int j = 0; j < NCHUNK; ++j) {
      bool skip = (used >> j) & 1u;
      float v = skip ? 3.4e38f : cv[j];
      if (v < bv || (v == bv && ci[j] < bidx)) { bv = v; bidx = ci[j]; bslot = j; }
    }
    int own = lane * NCHUNK + bslot;
    wave_argmin2(bv, bidx, own);
    if ((own >> 3) == lane) used |= 1u << (own & 7);
    if (lane == 0) {
      const float* p = xyz + ((size_t)(b * N_ + bidx) * 3);
      float* o = out_nb + (((b * G_ + gg) * K_ + k) * 3);
      o[0] = p[0] - cx;
      o[1] = p[1] - cy;
      o[2] = p[2] - cz;
    }
  }
}

// ---------------------------------------------------------------------------
extern "C" void kernel_launch(void* const* d_in, const int* in_sizes, int n_in,
                              void* d_out, int out_size, void* d_ws, size_t ws_size,
                              hipStream_t stream) {
  const float* pc   = (const float*)d_in[0];   // (8,16384,3) f32
  const float* pose = (const float*)d_in[1];   // (8,3,4)     f32

  float* xyz   = (float*)d_ws;                       // B*N*3
  float* pvals = xyz + B_ * N_ * 3;                  // B*G*NCHUNK*32
  int*   pidx  = (int*)(pvals + B_ * G_ * NCHUNK * K_);

  float* out_nb = (float*)d_out;                     // (8,512,32,3)
  float* out_ct = out_nb + B_ * G_ * K_ * 3;         // (8,512,3)

  se3_kernel<<<(B_ * N_) / 256, 256, 0, stream>>>(pc, pose, xyz);
  fps_kernel<<<B_, 1024, 0, stream>>>(xyz, out_ct);

  size_t smem = (size_t)(CHUNK * 3 + GT * 4 + GT * CHUNK) * sizeof(float); // ~152KB
  dist_topk_kernel<<<B_ * 32 * NCHUNK, 512, smem, stream>>>(xyz, out_ct, pvals, pidx);

  merge_kernel<<<(B_ * G_) / 8, 256, 0, stream>>>(xyz, out_ct, pvals, pidx, out_nb);
}